// GELU202_39857296507309
// MI455X (gfx1250) — compile-verified
//
#include <hip/hip_runtime.h>
#include <hip/hip_bf16.h>
#include <math.h>

// Problem constants (match reference): B=4, T=4096, D=2048, K=16
constexpr int DCH     = 2048;
constexpr int KSEL    = 16;
constexpr int NTHR    = 256;   // 8 wave32s
constexpr int NWAVES  = 8;

#if __has_builtin(__builtin_amdgcn_global_load_async_to_lds_b128)
#define USE_ASYNC 1
#else
#define USE_ASYNC 0
#endif

#if __has_builtin(__builtin_amdgcn_wmma_f32_16x16x4_f32)
#define USE_WMMA 1
#else
#define USE_WMMA 0
#endif

typedef float v2f __attribute__((ext_vector_type(2)));
typedef float v8f __attribute__((ext_vector_type(8)));
typedef int   v4i __attribute__((vector_size(16)));   // b128 payload type per builtin signature

#define GPTR(p, boff) ((__attribute__((address_space(1))) v4i*)((char*)(const char*)(p) + (boff)))
#define LPTR(p, boff) ((__attribute__((address_space(3))) v4i*)((char*)(p) + (boff)))

__device__ __forceinline__ float wave_max32(float v) {
#pragma unroll
  for (int m = 16; m >= 1; m >>= 1) v = fmaxf(v, __shfl_xor(v, m, 32));
  return v;
}
__device__ __forceinline__ float wave_sum32(float v) {
#pragma unroll
  for (int m = 16; m >= 1; m >>= 1) v += __shfl_xor(v, m, 32);
  return v;
}

__device__ __forceinline__ float gelu_tanh(float x) {
  const float c = 0.7978845608028654f;           // sqrt(2/pi)
  float x3 = x * x * x;
  return 0.5f * x * (1.0f + tanhf(c * (x + 0.044715f * x3)));
}

__global__ __launch_bounds__(NTHR)
void gelu_topk_gate_kernel(const float* __restrict__ x,
                           const float* __restrict__ ema_mean,
                           const float* __restrict__ ema_sq,
                           const float* __restrict__ ema_out,
                           const float* __restrict__ p_log_tau,
                           const float* __restrict__ p_log_bu,
                           const float* __restrict__ p_log_bd,
                           const float* __restrict__ p_log_g,
                           float* __restrict__ out) {
  __shared__ __align__(16) float S_mean[DCH];
  __shared__ __align__(16) float S_sq[DCH];
  __shared__ __align__(16) float S_out[DCH];
  __shared__ float Wc[NWAVES * KSEL];   // per-wave top-16 candidates
  __shared__ float A64[64];             // 16x4 A-matrix for WMMA merge
  __shared__ float bc[2];               // {gate_cos, threshold}

  const int  t     = threadIdx.x;
  const int  lane  = t & 31;
  const int  w     = t >> 5;
  const long base4 = (long)blockIdx.x * (DCH / 4);

  // ---- uniform scalar params (s_load, overlaps everything below) ----
  const float tau = expf(p_log_tau[0]);
  const float bu  = log1pf(expf(p_log_bu[0]));
  const float bd  = log1pf(expf(p_log_bd[0]));
  const float gam = log1pf(expf(p_log_g[0]));

  // ---- stage broadcast stats into LDS; overlap with x register loads ----
#if USE_ASYNC
  {
    const int off0 = t * 16;          // bytes 0..4095   (floats 0..1023)
    const int off1 = 4096 + t * 16;   // bytes 4096..8191 (floats 1024..2047)
    __builtin_amdgcn_global_load_async_to_lds_b128(GPTR(ema_mean, off0), LPTR(S_mean, off0), 0, 0);
    __builtin_amdgcn_global_load_async_to_lds_b128(GPTR(ema_mean, off1), LPTR(S_mean, off1), 0, 0);
    __builtin_amdgcn_global_load_async_to_lds_b128(GPTR(ema_sq,   off0), LPTR(S_sq,   off0), 0, 0);
    __builtin_amdgcn_global_load_async_to_lds_b128(GPTR(ema_sq,   off1), LPTR(S_sq,   off1), 0, 0);
    __builtin_amdgcn_global_load_async_to_lds_b128(GPTR(ema_out,  off0), LPTR(S_out,  off0), 0, 0);
    __builtin_amdgcn_global_load_async_to_lds_b128(GPTR(ema_out,  off1), LPTR(S_out,  off1), 0, 0);
  }
#else
  {
    const float4* m4 = (const float4*)ema_mean;
    const float4* q4 = (const float4*)ema_sq;
    const float4* e4 = (const float4*)ema_out;
    ((float4*)S_mean)[t]       = m4[t];
    ((float4*)S_mean)[t + 256] = m4[t + 256];
    ((float4*)S_sq)[t]         = q4[t];
    ((float4*)S_sq)[t + 256]   = q4[t + 256];
    ((float4*)S_out)[t]        = e4[t];
    ((float4*)S_out)[t + 256]  = e4[t + 256];
  }
#endif

  const float4* x4 = (const float4*)x;
  const float4  xa = x4[base4 + t];
  const float4  xb = x4[base4 + 256 + t];

#if USE_ASYNC
#if __has_builtin(__builtin_amdgcn_s_wait_asynccnt)
  __builtin_amdgcn_s_wait_asynccnt(0);
#else
  asm volatile("s_wait_asynccnt 0" ::: "memory");
#endif
#endif
  __syncthreads();   // LDS stats visible to all waves

  const float4 sm_a = ((const float4*)S_mean)[t];
  const float4 sm_b = ((const float4*)S_mean)[t + 256];
  const float4 sq_a = ((const float4*)S_sq)[t];
  const float4 sq_b = ((const float4*)S_sq)[t + 256];
  const float4 so_a = ((const float4*)S_out)[t];
  const float4 so_b = ((const float4*)S_out)[t + 256];

  float ox[8], zz[8], az[8];
  float ss = 0.f, dt = 0.f, se = 0.f;

#define PROC(i, XV, MV, QV, EV)                          \
  {                                                      \
    float o   = gelu_tanh(XV); ox[i] = o;                \
    float var = fmaxf((QV) - (MV) * (MV), 1e-4f);        \
    float zv  = ((XV) - (MV)) / (sqrtf(var) + 1e-5f);    \
    zz[i] = zv; az[i] = fabsf(zv);                       \
    ss += o * o; dt += o * (EV); se += (EV) * (EV);      \
  }
  PROC(0, xa.x, sm_a.x, sq_a.x, so_a.x)
  PROC(1, xa.y, sm_a.y, sq_a.y, so_a.y)
  PROC(2, xa.z, sm_a.z, sq_a.z, so_a.z)
  PROC(3, xa.w, sm_a.w, sq_a.w, so_a.w)
  PROC(4, xb.x, sm_b.x, sq_b.x, so_b.x)
  PROC(5, xb.y, sm_b.y, sq_b.y, so_b.y)
  PROC(6, xb.z, sm_b.z, sq_b.z, so_b.z)
  PROC(7, xb.w, sm_b.w, sq_b.w, so_b.w)
#undef PROC

  // ---- per-wave top-16 extraction over its 256 |z| values (shuffle-only) ----
  {
    float a0 = az[0], a1 = az[1], a2 = az[2], a3 = az[3];
    float a4 = az[4], a5 = az[5], a6 = az[6], a7 = az[7];
    for (int r = 0; r < KSEL; ++r) {
      float m  = fmaxf(fmaxf(fmaxf(a0, a1), fmaxf(a2, a3)),
                       fmaxf(fmaxf(a4, a5), fmaxf(a6, a7)));
      float wm = wave_max32(m);
      unsigned long long bal = __ballot(m == wm);
      if (m == wm && lane == (__ffsll(bal) - 1)) {
        if      (a0 == wm) a0 = -3.402823466e38f;
        else if (a1 == wm) a1 = -3.402823466e38f;
        else if (a2 == wm) a2 = -3.402823466e38f;
        else if (a3 == wm) a3 = -3.402823466e38f;
        else if (a4 == wm) a4 = -3.402823466e38f;
        else if (a5 == wm) a5 = -3.402823466e38f;
        else if (a6 == wm) a6 = -3.402823466e38f;
        else               a7 = -3.402823466e38f;
      }
      if (lane == 0) Wc[w * KSEL + r] = wm;
    }
  }

  // ---- per-wave sum reductions, deposit into 16x4 WMMA A-matrix ----
  ss = wave_sum32(ss);
  dt = wave_sum32(dt);
  se = wave_sum32(se);
  if (lane == 0) {            // rows 0-1: ss, rows 2-3: dt, rows 4-5: se
    A64[w]      = ss;         // indices 0..7
    A64[8 + w]  = dt;         // indices 8..15
    A64[16 + w] = se;         // indices 16..23
  }
  if (t >= 24 && t < 64) A64[t] = 0.f;  // zero-pad rest of A (disjoint indices)
  __syncthreads();

  // ---- wave 0: merge sums (matrix pipe) + merge top-k candidates ----
  if (w == 0) {
    float ssT, dtT, seT;
#if USE_WMMA
    v2f a;
    if (lane < 16) { a.x = A64[lane * 4 + 0];        a.y = A64[lane * 4 + 1]; }
    else           { a.x = A64[(lane - 16) * 4 + 2]; a.y = A64[(lane - 16) * 4 + 3]; }
    v2f b; b.x = 1.0f; b.y = 1.0f;                 // all-ones B => row sums
    v8f cc = {};
    cc = __builtin_amdgcn_wmma_f32_16x16x4_f32(false, a, false, b,
                                               (short)0, cc, false, false);
    ssT = cc[0] + cc[1];   // rows 0+1
    dtT = cc[2] + cc[3];   // rows 2+3
    seT = cc[4] + cc[5];   // rows 4+5 (valid on lanes 0..15)
#else
    float v = (lane < 24) ? A64[lane] : 0.f;
#pragma unroll
    for (int m = 4; m >= 1; m >>= 1) v += __shfl_xor(v, m, 32);
    ssT = __shfl(v, 0, 32); dtT = __shfl(v, 8, 32); seT = __shfl(v, 16, 32);
#endif
    if (lane == 0) {
      float cs = dtT / (fmaxf(sqrtf(ssT), 1e-12f) * fmaxf(sqrtf(seT), 1e-12f));
      cs = fminf(fmaxf(cs, -1.0f), 1.0f);
      bc[0] = expf(-tau * cs);
    }
    // merge 8x16 candidates -> 16th largest = threshold
    float c0 = Wc[lane * 4 + 0], c1 = Wc[lane * 4 + 1];
    float c2 = Wc[lane * 4 + 2], c3 = Wc[lane * 4 + 3];
    float th = 0.f;
    for (int r = 0; r < KSEL; ++r) {
      float m  = fmaxf(fmaxf(c0, c1), fmaxf(c2, c3));
      float wm = wave_max32(m);
      unsigned long long bal = __ballot(m == wm);
      if (m == wm && lane == (__ffsll(bal) - 1)) {
        if      (c0 == wm) c0 = -3.402823466e38f;
        else if (c1 == wm) c1 = -3.402823466e38f;
        else if (c2 == wm) c2 = -3.402823466e38f;
        else               c3 = -3.402823466e38f;
      }
      th = wm;
    }
    if (lane == 0) bc[1] = th;
  }
  __syncthreads();

  const float gcos = bc[0];
  const float th   = bc[1];

  float4 oa, ob;
#define FIN(i, DST)                                              \
  {                                                              \
    float g = 1.0f;                                              \
    if (az[i] >= th) {                                           \
      float tz = tanhf(gam * zz[i]);                             \
      g = 1.0f + bu * fmaxf(tz, 0.f) - bd * fmaxf(-tz, 0.f);     \
      g = fminf(fmaxf(g, 0.05f), 8.0f);                          \
    }                                                            \
    DST = ox[i] * g * gcos;                                      \
  }
  FIN(0, oa.x) FIN(1, oa.y) FIN(2, oa.z) FIN(3, oa.w)
  FIN(4, ob.x) FIN(5, ob.y) FIN(6, ob.z) FIN(7, ob.w)
#undef FIN

  float4* o4 = (float4*)out;
  o4[base4 + t]       = oa;
  o4[base4 + 256 + t] = ob;
}

extern "C" void kernel_launch(void* const* d_in, const int* in_sizes, int n_in,
                              void* d_out, int out_size, void* d_ws, size_t ws_size,
                              hipStream_t stream) {
  const float* x        = (const float*)d_in[0];
  const float* ema_mean = (const float*)d_in[1];
  const float* ema_sq   = (const float*)d_in[2];
  const float* ema_out  = (const float*)d_in[3];
  const float* log_tau  = (const float*)d_in[4];
  const float* log_bu   = (const float*)d_in[5];
  const float* log_bd   = (const float*)d_in[6];
  const float* log_g    = (const float*)d_in[7];
  float* out = (float*)d_out;

  const int tokens = in_sizes[0] / DCH;   // B*T = 16384
  gelu_topk_gate_kernel<<<dim3(tokens), dim3(NTHR), 0, stream>>>(
      x, ema_mean, ema_sq, ema_out, log_tau, log_bu, log_bd, log_g, out);
}